// vlad_52475910422535
// MI455X (gfx1250) — compile-verified
//
#include <hip/hip_runtime.h>
#include <math.h>

#define N_PTS   4096
#define K_CENT  125
#define K_PAD   128
#define DIM     512
#define PD_EPS  1e-6f
#define NORM_EPS 1e-12f

typedef __attribute__((ext_vector_type(2))) float v2f;
typedef __attribute__((ext_vector_type(4))) float v4f;
typedef __attribute__((ext_vector_type(8))) float v8f;

// ---------------------------------------------------------------------------
// Kernel 1: per-row sum and squared-norm (one wave32 per row)
// ---------------------------------------------------------------------------
__global__ void vlad_rowstats(const float* __restrict__ M, int D,
                              float* __restrict__ norms, float* __restrict__ sums) {
    const int row  = blockIdx.x;
    const int lane = threadIdx.x;                 // 0..31
    const float* p = M + (size_t)row * D;
    float s = 0.f, nrm = 0.f;
    for (int d = lane; d < D; d += 32) {
        float v = p[d];
        s   += v;
        nrm += v * v;
    }
    for (int off = 16; off > 0; off >>= 1) {
        s   += __shfl_down(s,   off, 32);
        nrm += __shfl_down(nrm, off, 32);
    }
    if (lane == 0) { norms[row] = nrm; sums[row] = s; }
}

// ---------------------------------------------------------------------------
// Kernel 2: S = X * C^T via V_WMMA_F32_16X16X4_F32 (one wave per 16x16 tile)
//   A (16x4 f32): lanes 0-15 hold M=lane, {K=0,K=1}; lanes 16-31 hold {K=2,K=3}
//   B (4x16 f32): lanes 0-15 hold N=lane, {K=0,K=1}; lanes 16-31 hold {K=2,K=3}
//   C/D (16x16 f32): VGPR r: lanes 0-15 -> M=r, N=lane; lanes 16-31 -> M=8+r
// ---------------------------------------------------------------------------
__global__ void vlad_gemm_xc(const float* __restrict__ X,
                             const float* __restrict__ C,
                             float* __restrict__ S) {
    const int lane  = threadIdx.x;                // 0..31
    const int mBase = blockIdx.x * 16;            // 256 tiles over 4096 rows
    const int nBase = blockIdx.y * 16;            // 8 tiles over 128 padded cols
    const int mRow  = mBase + (lane & 15);
    const int nCol  = nBase + (lane & 15);
    const int kOff  = (lane >> 4) * 2;            // 0 or 2
    const bool colOK = (nCol < K_CENT);
    const int nColC  = colOK ? nCol : (K_CENT - 1);   // clamp pointer, mask value

    const float* xp = X + (size_t)mRow  * DIM + kOff;
    const float* cp = C + (size_t)nColC * DIM + kOff;

    v8f acc = {};
#pragma unroll 4
    for (int k = 0; k < DIM; k += 4) {
        v2f a = *(const v2f*)(xp + k);
        v2f b = *(const v2f*)(cp + k);
        b.x = colOK ? b.x : 0.f;                  // v_cndmask, no EXEC change
        b.y = colOK ? b.y : 0.f;
        acc = __builtin_amdgcn_wmma_f32_16x16x4_f32(
                  /*neg_a=*/false, a, /*neg_b=*/false, b,
                  /*c_mod=*/(short)0, acc, /*reuse_a=*/false, /*reuse_b=*/false);
    }

    const int col    = nBase + (lane & 15);
    const int rowOff = (lane >> 4) * 8;
#pragma unroll
    for (int r = 0; r < 8; ++r) {
        S[(size_t)(mBase + rowOff + r) * K_PAD + col] = acc[r];
    }
}

// ---------------------------------------------------------------------------
// Kernel 3: per-row softmax over 1/dist and fused scale; overwrites S in place
// ---------------------------------------------------------------------------
__device__ __forceinline__ float blockMax128(float v, float* red, int k) {
    red[k] = v; __syncthreads();
    for (int s = 64; s > 0; s >>= 1) {
        if (k < s) red[k] = fmaxf(red[k], red[k + s]);
        __syncthreads();
    }
    float r = red[0]; __syncthreads();
    return r;
}
__device__ __forceinline__ float blockSum128(float v, float* red, int k) {
    red[k] = v; __syncthreads();
    for (int s = 64; s > 0; s >>= 1) {
        if (k < s) red[k] = red[k] + red[k + s];
        __syncthreads();
    }
    float r = red[0]; __syncthreads();
    return r;
}

__global__ void vlad_softmax_scale(float* __restrict__ S,
                                   const float* __restrict__ normX,
                                   const float* __restrict__ sumX,
                                   const float* __restrict__ normC,
                                   const float* __restrict__ sumC) {
    __shared__ float red[K_PAD];
    const int n  = blockIdx.x;
    const int k  = threadIdx.x;                   // 0..127
    const bool ok = (k < K_CENT);

    const float dot = S[(size_t)n * K_PAD + k];
    const float nx  = normX[n], sx = sumX[n];
    const float nc  = ok ? normC[k] : 0.f;
    const float sc  = ok ? sumC[k]  : 0.f;

    const float d2      = nx + nc - 2.f * dot;                     // ||x-c||^2
    const float distArg = d2 + 2.f * PD_EPS * (sx - sc)
                             + (float)DIM * PD_EPS * PD_EPS;
    const float dist    = sqrtf(fmaxf(distArg, 0.f));
    const float resNorm = sqrtf(fmaxf(d2, 0.f));

    const float inv  = ok ? (1.f / dist) : -3.402823466e38f;
    const float mx   = blockMax128(inv, red, k);
    const float e    = ok ? expf(inv - mx) : 0.f;
    const float sumE = blockSum128(e, red, k);
    const float assign = e / sumE;

    const float t   = assign * resNorm;           // ||assign * resid||
    const float tm  = fmaxf(t, NORM_EPS);
    const float u   = t / tm;                     // intra-normalized block norm
    const float rowSq = blockSum128(ok ? u * u : 0.f, red, k);
    const float rowF  = fmaxf(sqrtf(rowSq), NORM_EPS);

    const float scale = assign / (tm * rowF);
    if (ok) S[(size_t)n * K_PAD + k] = scale;
}

// ---------------------------------------------------------------------------
// Kernel 4: streaming output (bandwidth-bound): out = (x - c) * scale, NT stores
// ---------------------------------------------------------------------------
__global__ void vlad_write(const float* __restrict__ X,
                           const float* __restrict__ C,
                           const float* __restrict__ scaleBuf,
                           float* __restrict__ out) {
    const int kc = blockIdx.x;                    // 0..124
    const int n  = blockIdx.y;                    // 0..4095
    const int t  = threadIdx.x;                   // 0..127 (x4 floats = 512)

    const float s = scaleBuf[(size_t)n * K_PAD + kc];
    const v4f xv = ((const v4f*)(X + (size_t)n  * DIM))[t];
    const v4f cv = ((const v4f*)(C + (size_t)kc * DIM))[t];
    v4f r = (xv - cv) * s;

    v4f* dst = (v4f*)(out + (size_t)n * ((size_t)K_CENT * DIM) + (size_t)kc * DIM) + t;
    __builtin_nontemporal_store(r, dst);
}

// ---------------------------------------------------------------------------
extern "C" void kernel_launch(void* const* d_in, const int* in_sizes, int n_in,
                              void* d_out, int out_size, void* d_ws, size_t ws_size,
                              hipStream_t stream) {
    const float* X = (const float*)d_in[0];       // 4096 x 512
    const float* C = (const float*)d_in[1];       // 125  x 512
    float* out = (float*)d_out;                   // 4096 x 64000

    // workspace layout (~2.03 MB)
    float* S     = (float*)d_ws;                  // 4096 x 128 (dots -> scales)
    float* normX = S + (size_t)N_PTS * K_PAD;
    float* sumX  = normX + N_PTS;
    float* normC = sumX + N_PTS;
    float* sumC  = normC + K_PAD;

    vlad_rowstats<<<N_PTS, 32, 0, stream>>>(X, DIM, normX, sumX);
    vlad_rowstats<<<K_CENT, 32, 0, stream>>>(C, DIM, normC, sumC);
    vlad_gemm_xc<<<dim3(N_PTS / 16, K_PAD / 16), 32, 0, stream>>>(X, C, S);
    vlad_softmax_scale<<<N_PTS, K_PAD, 0, stream>>>(S, normX, sumX, normC, sumC);
    vlad_write<<<dim3(K_CENT, N_PTS), 128, 0, stream>>>(X, C, S, out);
}